// PatchWiseProduct_22308060135555
// MI455X (gfx1250) — compile-verified
//
#include <hip/hip_runtime.h>

typedef float v2f __attribute__((ext_vector_type(2)));
typedef float v8f __attribute__((ext_vector_type(8)));

#define BATCH 16
#define H_IN  256
#define W_IN  256
#define C_IN  16
#define H_OUT 128
#define W_OUT 128
#define C_OUT 256
#define PITCH 17   // floats per (h,col) channel row in LDS; odd -> conflict-free strided reads

// out[b,i,j,o] = sum_{r,s} x[b, 2i+r, 2j+s, k[r,s,o]]
// Cast as GEMM: A[p, K=64] (patch taps x channels, fp32) times one-hot W[64, 256],
// computed exactly with V_WMMA_F32_16X16X4_F32 (16 K-steps of 4).
__global__ __launch_bounds__(256)
void PatchWiseProduct_wmma_kernel(const float* __restrict__ x,
                                  const int* __restrict__ sk,   // [2][2][256] -> tap-major [4][256]
                                  float* __restrict__ out)
{
    // Two input rows (h = 2i, 2i+1), 256 cols, 16 channels, pitch-padded.
    __shared__ float xs[2 * W_IN * PITCH];   // 34,816 B

    const int tid = threadIdx.x;
    const int wg  = blockIdx.x;            // b*H_OUT + i
    const int b   = wg >> 7;
    const int i   = wg & 127;

    // ---- stage x rows 2i and 2i+1 into LDS (coalesced float4 global loads) ----
    const long rowbase0 = ((long)(b * H_IN + 2 * i) * W_IN) * C_IN;
    const long rowbase1 = rowbase0 + (long)W_IN * C_IN;
    for (int idx = tid; idx < 2048; idx += 256) {      // 2048 float4 chunks
        const int h   = idx >> 10;                     // 0..1
        const int rem = idx & 1023;                    // float4 index within row
        const float4 v = *(const float4*)(x + (h ? rowbase1 : rowbase0) + (rem << 2));
        const int col = rem >> 2;                      // 0..255
        const int c   = (rem & 3) << 2;                // 0,4,8,12
        float* dst = &xs[(h * W_IN + col) * PITCH + c];
        dst[0] = v.x; dst[1] = v.y; dst[2] = v.z; dst[3] = v.w;
    }
    __syncthreads();

    const int wave = tid >> 5;       // 0..7 : patch tile  j = wave*16 + m
    const int lane = tid & 31;
    const int m    = lane & 15;      // M row (and N col) index within 16x16 tile
    const int half = lane >> 4;      // 0/1 : K split across lane halves

    // ---- A fragments: a[kk] holds A[m][k], k = kk*4 + half*2 + {0,1} ----
    // k -> tap t = k>>4 (r = t>>1, s = t&1), channel c = k&15
    v2f a[16];
#pragma unroll
    for (int kk = 0; kk < 16; ++kk) {
        const int t  = kk >> 2;                         // tap, constant over lanes
        const int r  = t >> 1, s = t & 1;
        const int c0 = ((kk & 3) << 2) + (half << 1);   // channel of component .x
        const int col = (((wave << 4) + m) << 1) + s;   // 2*j + s
        const float* p = &xs[(r * W_IN + col) * PITCH + c0];
        a[kk].x = p[0];
        a[kk].y = p[1];
    }

    const long obase = ((long)wg * W_OUT) * C_OUT;      // flat base of (b,i,*,*)

#pragma unroll 1
    for (int nt = 0; nt < 16; ++nt) {                   // output-channel tiles
        const int oc = (nt << 4) + m;                   // this lane's N column
        const int s0 = sk[0 * C_OUT + oc];
        const int s1 = sk[1 * C_OUT + oc];
        const int s2 = sk[2 * C_OUT + oc];
        const int s3 = sk[3 * C_OUT + oc];
        v8f acc = {};
#pragma unroll
        for (int kk = 0; kk < 16; ++kk) {               // K = 64 in steps of 4
            const int st = (kk < 4) ? s0 : (kk < 8) ? s1 : (kk < 12) ? s2 : s3;
            const int c0 = ((kk & 3) << 2) + (half << 1);
            v2f bf;                                     // one-hot B fragment
            bf.x = (st == c0    ) ? 1.0f : 0.0f;
            bf.y = (st == c0 + 1) ? 1.0f : 0.0f;
            // D = A x B + C, fp32-exact
            acc = __builtin_amdgcn_wmma_f32_16x16x4_f32(
                false, a[kk], false, bf, (short)0, acc, false, false);
        }
        // ---- store 16x16 f32 D tile: VGPR v -> M = v + 8*half, N = lane&15 ----
#pragma unroll
        for (int v = 0; v < 8; ++v) {
            const int M = v + (half << 3);
            const int j = (wave << 4) + M;
            out[obase + (long)j * C_OUT + oc] = acc[v];
        }
    }
}

extern "C" void kernel_launch(void* const* d_in, const int* in_sizes, int n_in,
                              void* d_out, int out_size, void* d_ws, size_t ws_size,
                              hipStream_t stream) {
    const float* x  = (const float*)d_in[0];
    const int*   sk = (const int*)d_in[1];
    float*       out = (float*)d_out;
    (void)in_sizes; (void)n_in; (void)out_size; (void)d_ws; (void)ws_size;

    dim3 grid(BATCH * H_OUT);   // 2048 workgroups, one per (b, i) output row
    dim3 block(256);            // 8 waves of 32
    PatchWiseProduct_wmma_kernel<<<grid, block, 0, stream>>>(x, sk, out);
}